// STGCN_Residual_24043226923274
// MI455X (gfx1250) — compile-verified
//
#include <hip/hip_runtime.h>
#include <math.h>

typedef __attribute__((ext_vector_type(2))) float v2f;
typedef __attribute__((ext_vector_type(8))) float v8f;

#define N_NODES 10000
#define SEQ_LEN 12
#define IN_CH   16
#define HID     64
#define N_EDGES 160000
#define NT      (N_NODES * SEQ_LEN)   // 120000 rows

// ---------------------------------------------------------------------------
// fp32 WMMA: D(16x16) = A(16x4) * B(4x16) + C  — V_WMMA_F32_16X16X4_F32
// A layout: lane L holds M=L%16; v0 -> K = (L<16?0:2), v1 -> K+1  (per k-block)
// B layout: lane L holds N=L%16; v0 -> K = (L<16?0:2), v1 -> K+1
// C layout: VGPR i, lanes 0-15: M=i, lanes 16-31: M=i+8; N = L%16
// ---------------------------------------------------------------------------
static __device__ __forceinline__ v8f wmma4(v2f a, v2f b, v8f c) {
  return __builtin_amdgcn_wmma_f32_16x16x4_f32(false, a, false, b, (short)0, c,
                                               false, false);
}
static __device__ __forceinline__ float sigm(float x) {
  return 1.0f / (1.0f + __expf(-x));
}

// ---------------------------------------------------------------------------
// 0) init: deg = 1 (self loop), agg = 0
// ---------------------------------------------------------------------------
__global__ void k_init(float* deg, float* agg, int nAgg) {
  int i = blockIdx.x * blockDim.x + threadIdx.x;
  if (i < nAgg) agg[i] = 0.0f;
  if (i < N_NODES) deg[i] = 1.0f;
}

// 1) degree accumulation over dst
__global__ void k_deg(const int* __restrict__ ei, float* deg) {
  int e = blockIdx.x * blockDim.x + threadIdx.x;
  if (e < N_EDGES) unsafeAtomicAdd(&deg[ei[N_EDGES + e]], 1.0f);
}

// 2) dis = rsqrt(deg), in place
__global__ void k_rsqrt(float* deg) {
  int i = blockIdx.x * blockDim.x + threadIdx.x;
  if (i < N_NODES) deg[i] = rsqrtf(deg[i]);
}

// 3) edge message scatter: agg[dst,t,:] += x[src,t,:] * dis[src]*dis[dst]
__global__ void k_scatter(const int* __restrict__ ei, const float* __restrict__ x,
                          const float* __restrict__ dis, float* agg) {
  int tid = blockIdx.x * blockDim.x + threadIdx.x;
  if (tid >= N_EDGES * SEQ_LEN) return;
  int e = tid / SEQ_LEN, t = tid % SEQ_LEN;
  int s = ei[e], d = ei[N_EDGES + e];
  float nrm = dis[s] * dis[d];
  const float4* xs = (const float4*)(x + (size_t)t * N_NODES * IN_CH + (size_t)s * IN_CH);
  float* ap = agg + ((size_t)d * SEQ_LEN + t) * IN_CH;
#pragma unroll
  for (int q = 0; q < 4; ++q) {
    float4 v = xs[q];
    unsafeAtomicAdd(ap + 4 * q + 0, v.x * nrm);
    unsafeAtomicAdd(ap + 4 * q + 1, v.y * nrm);
    unsafeAtomicAdd(ap + 4 * q + 2, v.z * nrm);
    unsafeAtomicAdd(ap + 4 * q + 3, v.w * nrm);
  }
}

// ---------------------------------------------------------------------------
// 4) GCN GEMM + self loop + bias + ReLU:  out(120000x64) = relu(A(.x16)*W(16x64)+b)
//    1 wave = one 16-row tile x all 64 cols (4 col tiles), K=16 -> 4 wmma k-steps.
//    7500 tiles = 1875 blocks * 4 waves, exact (EXEC stays all-ones for WMMA).
// ---------------------------------------------------------------------------
__global__ void __launch_bounds__(128) k_gcn(const float* __restrict__ agg,
                                             const float* __restrict__ x,
                                             const float* __restrict__ dis,
                                             const float* __restrict__ w,
                                             const float* __restrict__ b,
                                             float* __restrict__ out) {
  int lane  = threadIdx.x & 31;
  int wv    = threadIdx.x >> 5;
  int tile  = blockIdx.x * 4 + wv;
  int col16 = lane & 15;
  int khalf = (lane < 16) ? 0 : 2;

  int R0 = tile * 16;
  int R  = R0 + col16;                 // A-operand row for this lane
  int n  = R / SEQ_LEN, t = R % SEQ_LEN;
  float dn = dis[n];
  float selfn = dn * dn;
  const float* aRow = agg + (size_t)R * IN_CH + khalf;
  const float* xRow = x + (size_t)t * N_NODES * IN_CH + (size_t)n * IN_CH + khalf;

  v8f zero = {};
  v8f acc[4];
#pragma unroll
  for (int c = 0; c < 4; ++c) acc[c] = zero;

#pragma unroll
  for (int kk = 0; kk < 4; ++kk) {
    float2 ag = *(const float2*)(aRow + 4 * kk);
    float2 xv = *(const float2*)(xRow + 4 * kk);
    v2f a;
    a.x = ag.x + selfn * xv.x;         // fold self-loop into A
    a.y = ag.y + selfn * xv.y;
#pragma unroll
    for (int c = 0; c < 4; ++c) {
      int colg = c * 16 + col16;
      v2f bb;
      bb.x = w[(4 * kk + khalf) * HID + colg];
      bb.y = w[(4 * kk + khalf + 1) * HID + colg];
      acc[c] = wmma4(a, bb, acc[c]);
    }
  }

#pragma unroll
  for (int c = 0; c < 4; ++c) {
    int colg = c * 16 + col16;
    float bias = b[colg];
#pragma unroll
    for (int i = 0; i < 8; ++i) {
      int row = (lane < 16) ? i : i + 8;
      float v = acc[c][i] + bias;
      out[(size_t)(R0 + row) * HID + colg] = fmaxf(v, 0.0f);
    }
  }
}

// ---------------------------------------------------------------------------
// 5/6) GRU layer scan. Recurrence is per-node -> one wave owns 16 nodes for all
//      12 timesteps; h lives in the wave's private LDS slice. Per t, both
//      x_t*W_ih^T and h*W_hh^T run on WMMA; r/z columns share an accumulator
//      (C accumulates both GEMMs), n-gate keeps x/h parts separate.
//      625 waves = 125 blocks * 5 waves, exact.
// ---------------------------------------------------------------------------
__global__ void __launch_bounds__(160) k_gru(const float* __restrict__ inseq,
                                             const float* __restrict__ w_ih,
                                             const float* __restrict__ w_hh,
                                             const float* __restrict__ b_ih,
                                             const float* __restrict__ b_hh,
                                             float* __restrict__ outseq) {
  __shared__ float hbuf[5][16][HID];   // 20 KB: per-wave h tile (16 nodes x 64)
  int lane  = threadIdx.x & 31;
  int wv    = threadIdx.x >> 5;
  int col16 = lane & 15;
  int khalf = (lane < 16) ? 0 : 2;
  int nodeBase = (blockIdx.x * 5 + wv) * 16;

  // h0 = 0 (each wave zeroes only its own slice; same-wave LDS is in-order)
  float* hflat = &hbuf[wv][0][0];
#pragma unroll
  for (int i = 0; i < 32; ++i) hflat[lane + 32 * i] = 0.0f;
  __builtin_amdgcn_wave_barrier();

  // hoisted per-lane biases (j = gate column for this lane in col-tile c)
  float biasR[4], biasZ[4], bInX[4], bInH[4];
#pragma unroll
  for (int c = 0; c < 4; ++c) {
    int j = c * 16 + col16;
    biasR[c] = b_ih[j] + b_hh[j];
    biasZ[c] = b_ih[64 + j] + b_hh[64 + j];
    bInX[c]  = b_ih[128 + j];
    bInH[c]  = b_hh[128 + j];
  }

  // B operand bases: B[k][j] = w[j*64 + k]  (x @ W^T), pairs (k,k+1) contiguous
  const float* bihB[12];
  const float* bhhB[12];
#pragma unroll
  for (int c = 0; c < 12; ++c) {
    int j = c * 16 + col16;
    bihB[c] = w_ih + j * HID + khalf;
    bhhB[c] = w_hh + j * HID + khalf;
  }
  const float* aRow0 =
      inseq + (size_t)(nodeBase + col16) * SEQ_LEN * HID + khalf;
  v8f zero = {};

  for (int t = 0; t < SEQ_LEN; ++t) {
    const float* aRow = aRow0 + (size_t)t * HID;
    v8f accRZ[8], accNX[4], accNH[4];
#pragma unroll
    for (int c = 0; c < 8; ++c) accRZ[c] = zero;
#pragma unroll
    for (int c = 0; c < 4; ++c) { accNX[c] = zero; accNH[c] = zero; }

#pragma unroll 4
    for (int kk = 0; kk < 16; ++kk) {
      v2f ax, ah;
      { float2 tmp = *(const float2*)(aRow + 4 * kk); ax.x = tmp.x; ax.y = tmp.y; }
      { float2 tmp = *(const float2*)(&hbuf[wv][col16][4 * kk + khalf]);
        ah.x = tmp.x; ah.y = tmp.y; }
#pragma unroll
      for (int c = 0; c < 8; ++c) {           // r,z gate columns: shared chain
        v2f bx, bh;
        { float2 tmp = *(const float2*)(bihB[c] + 4 * kk); bx.x = tmp.x; bx.y = tmp.y; }
        accRZ[c] = wmma4(ax, bx, accRZ[c]);
        { float2 tmp = *(const float2*)(bhhB[c] + 4 * kk); bh.x = tmp.x; bh.y = tmp.y; }
        accRZ[c] = wmma4(ah, bh, accRZ[c]);
      }
#pragma unroll
      for (int c = 0; c < 4; ++c) {           // n gate: keep x / h parts apart
        v2f bx, bh;
        { float2 tmp = *(const float2*)(bihB[8 + c] + 4 * kk); bx.x = tmp.x; bx.y = tmp.y; }
        accNX[c] = wmma4(ax, bx, accNX[c]);
        { float2 tmp = *(const float2*)(bhhB[8 + c] + 4 * kk); bh.x = tmp.x; bh.y = tmp.y; }
        accNH[c] = wmma4(ah, bh, accNH[c]);
      }
    }

    // gates + h update (C layout: vgpr i -> row i or i+8; col = c*16 + lane%16)
#pragma unroll
    for (int c = 0; c < 4; ++c) {
      int j = c * 16 + col16;
#pragma unroll
      for (int i = 0; i < 8; ++i) {
        int row = (lane < 16) ? i : i + 8;
        float r  = sigm(accRZ[c][i] + biasR[c]);
        float z  = sigm(accRZ[c + 4][i] + biasZ[c]);
        float nn = tanhf(accNX[c][i] + bInX[c] + r * (accNH[c][i] + bInH[c]));
        float ho = hbuf[wv][row][j];
        float hn = (1.0f - z) * nn + z * ho;
        hbuf[wv][row][j] = hn;
        outseq[((size_t)(nodeBase + row) * SEQ_LEN + t) * HID + j] = hn;
      }
    }
    __builtin_amdgcn_wave_barrier();   // pin LDS write->read ordering per wave
  }
}

// ---------------------------------------------------------------------------
// 7) attention softmax over T + FC + residual.
//    sum_h ctx[h]*fc_w[h] == sum_t p[t] * (g_t . fc_w) -> two dots per t.
// ---------------------------------------------------------------------------
__global__ void k_attn(const float* __restrict__ h2, const float* __restrict__ x,
                       const float* __restrict__ attn_w, const float* __restrict__ attn_b,
                       const float* __restrict__ fc_w, const float* __restrict__ fc_b,
                       float* __restrict__ out) {
  int n = blockIdx.x * blockDim.x + threadIdx.x;
  if (n >= N_NODES) return;
  const float* g = h2 + (size_t)n * SEQ_LEN * HID;
  float s[SEQ_LEN], d2[SEQ_LEN];
  float m = -1e30f;
  for (int t = 0; t < SEQ_LEN; ++t) {
    float dA = 0.0f, dF = 0.0f;
#pragma unroll
    for (int h = 0; h < HID; ++h) {
      float gv = g[t * HID + h];
      dA += gv * attn_w[h];
      dF += gv * fc_w[h];
    }
    s[t] = dA + attn_b[0];
    d2[t] = dF;
    m = fmaxf(m, s[t]);
  }
  float den = 0.0f;
#pragma unroll
  for (int t = 0; t < SEQ_LEN; ++t) { s[t] = __expf(s[t] - m); den += s[t]; }
  float inv = 1.0f / den, res = 0.0f;
#pragma unroll
  for (int t = 0; t < SEQ_LEN; ++t) res += s[t] * inv * d2[t];
  res += fc_b[0];
  out[n] = x[(size_t)(SEQ_LEN - 1) * N_NODES * IN_CH + (size_t)n * IN_CH] + res;
}

// ---------------------------------------------------------------------------
extern "C" void kernel_launch(void* const* d_in, const int* in_sizes, int n_in,
                              void* d_out, int out_size, void* d_ws, size_t ws_size,
                              hipStream_t stream) {
  const float* x      = (const float*)d_in[0];
  const int*   ei     = (const int*)d_in[1];
  const float* gcn_w  = (const float*)d_in[2];
  const float* gcn_b  = (const float*)d_in[3];
  const float* w_ih0  = (const float*)d_in[4];
  const float* w_hh0  = (const float*)d_in[5];
  const float* b_ih0  = (const float*)d_in[6];
  const float* b_hh0  = (const float*)d_in[7];
  const float* w_ih1  = (const float*)d_in[8];
  const float* w_hh1  = (const float*)d_in[9];
  const float* b_ih1  = (const float*)d_in[10];
  const float* b_hh1  = (const float*)d_in[11];
  const float* attn_w = (const float*)d_in[12];
  const float* attn_b = (const float*)d_in[13];
  const float* fc_w   = (const float*)d_in[14];
  const float* fc_b   = (const float*)d_in[15];

  float* ws      = (float*)d_ws;
  float* deg     = ws;                        // 10000 floats (dis in place)
  float* agg     = ws + 10240;                // 1,920,000 floats
  float* gcn_out = agg + (size_t)NT * IN_CH;  // 7,680,000 floats
  float* h1      = gcn_out + (size_t)NT * HID;// 7,680,000 floats
  float* h2      = gcn_out;                   // reuse: gcn_out dead after layer0

  const int nAgg = NT * IN_CH;
  k_init<<<(nAgg + 255) / 256, 256, 0, stream>>>(deg, agg, nAgg);
  k_deg<<<(N_EDGES + 255) / 256, 256, 0, stream>>>(ei, deg);
  k_rsqrt<<<(N_NODES + 255) / 256, 256, 0, stream>>>(deg);
  k_scatter<<<(N_EDGES * SEQ_LEN + 255) / 256, 256, 0, stream>>>(ei, x, deg, agg);
  k_gcn<<<NT / 16 / 4, 128, 0, stream>>>(agg, x, deg, gcn_w, gcn_b, gcn_out);
  k_gru<<<N_NODES / 16 / 5, 160, 0, stream>>>(gcn_out, w_ih0, w_hh0, b_ih0, b_hh0, h1);
  k_gru<<<N_NODES / 16 / 5, 160, 0, stream>>>(h1, w_ih1, w_hh1, b_ih1, b_hh1, h2);
  k_attn<<<(N_NODES + 255) / 256, 256, 0, stream>>>(h2, x, attn_w, attn_b, fc_w,
                                                    fc_b, (float*)d_out);
}